// loTeNet_69930657513827
// MI455X (gfx1250) — compile-verified
//
#include <hip/hip_runtime.h>
#include <hip/hip_bf16.h>
#include <math.h>

typedef __attribute__((ext_vector_type(2))) float v2f;
typedef __attribute__((ext_vector_type(4))) float v4f;
typedef __attribute__((ext_vector_type(8))) float v8f;

#define PI_HALF 1.57079632679489662f

// ---------------------------------------------------------------------------
// prep0: squeeze(x, 4) -> raw site values v0[b, n, s]  (B=128, N=256, S=16)
// also zero layer-0 BN stats
// ---------------------------------------------------------------------------
__global__ void prep0_kernel(const float* __restrict__ x, float* __restrict__ v0,
                             float* __restrict__ st0) {
  int i = blockIdx.x * blockDim.x + threadIdx.x;
  if (i < 128 * 256 * 16) {
    int s = i & 15, n = (i >> 4) & 255, b = i >> 12;
    int hb = n >> 4, wb = n & 15, dh = s >> 2, dw = s & 3;
    v0[i] = x[(b * 64 + hb * 4 + dh) * 64 + (wb * 4 + dw)];
  }
  if (i < 4) st0[i] = 0.0f;
}

// ---------------------------------------------------------------------------
// prep1: BN(y0) with batch stats (memory-reinterpret channel = p>>8),
// reinterpret (B,16,16,V), squeeze(4) -> v1[b, n1, s1]; zero layer-1 stats
// ---------------------------------------------------------------------------
__global__ void prep1_kernel(const float* __restrict__ y0, const float* __restrict__ st0,
                             const float* __restrict__ g, const float* __restrict__ be,
                             float* __restrict__ v1, float* __restrict__ st1) {
  int i = blockIdx.x * blockDim.x + threadIdx.x;
  if (i < 128 * 512) {
    int p = i & 511, ch = p >> 8;
    const float cnt = 128.0f * 256.0f;
    float mean = st0[ch] / cnt;
    float var  = st0[2 + ch] / cnt - mean * mean;
    float z = (y0[i] - mean) * rsqrtf(var + 1e-5f) * g[ch] + be[ch];
    int b = i >> 9;
    int hp = p >> 5, wp = (p >> 1) & 15, vv = p & 1;
    int n1 = (hp >> 2) * 4 + (wp >> 2);
    int s1 = ((hp & 3) * 4 + (wp & 3)) * 2 + vv;
    v1[(b * 16 + n1) * 32 + s1] = z;
  }
  if (i < 4) st1[i] = 0.0f;
}

// ---------------------------------------------------------------------------
// prep2: BN(y1) (channel = p>>4), reinterpret (B,4,4,V), squeeze(4) -> v2[b,s]
// ---------------------------------------------------------------------------
__global__ void prep2_kernel(const float* __restrict__ y1, const float* __restrict__ st1,
                             const float* __restrict__ g, const float* __restrict__ be,
                             float* __restrict__ v2) {
  int i = blockIdx.x * blockDim.x + threadIdx.x;
  if (i < 128 * 32) {
    int p = i & 31, ch = p >> 4;
    const float cnt = 128.0f * 16.0f;
    float mean = st1[ch] / cnt;
    float var  = st1[2 + ch] / cnt - mean * mean;
    float z = (y1[i] - mean) * rsqrtf(var + 1e-5f) * g[ch] + be[ch];
    int b = i >> 5;
    int h = p >> 3, w = (p >> 1) & 3, vv = p & 1;
    int s = (h * 4 + w) * 2 + vv;
    v2[b * 32 + s] = z;
  }
}

// ---------------------------------------------------------------------------
// Batched MPO chain, one pixel per block, deferred-scaling formulation.
//
// State kept in LDS as the raw WMMA output pair u = c @ [Wm0 | Wm1],
// stored TRANSPOSED (uBuf[tile][k][m]) so each lane's 8 accumulator values
// go out as two ds_store_b128. The per-row (cos,sin) embedding scale is
// applied while building the next A operand: in A layout each lane owns a
// single row m = lane&15, so the scale is one per-lane float2.
//
//   step j:  A[m][k] = u0[k][m]*cos(site j, row m) + u1[k][m]*sin(...)
//            [u0' | u1'] = A @ [Wm0_j | Wm1_j]        (2x V_WMMA_F32_16X16X4)
// ---------------------------------------------------------------------------
template <int O, int BPB, int S, int NMID, int NPIX, bool FINAL>
__global__ __launch_bounds__(128) void mpo_chain_kernel(
    const float* __restrict__ v,   // [128, NPIX, S] raw site values
    const float* __restrict__ Wf,  // [NPIX, 2, O, 8]
    const float* __restrict__ Wm,  // [NPIX, NMID, 8, 2, 8]
    const float* __restrict__ Wl,  // [NPIX, 8, 2]
    float* __restrict__ y,         // [128, NPIX, O], or [128, O] if FINAL
    float* __restrict__ stats,     // {sum0, sum1, sq0, sq1} (unused if FINAL)
    int half_thr) {
  constexpr int ROWS = BPB * O;
  constexpr int TILES = ROWS / 16;
  __shared__ float uBuf[TILES][16][16];    // [tile][k-col][m]  (transposed state)
  __shared__ float csv[BPB][S][2];         // per-batch per-site (cos, sin)
  __shared__ float BmP[NMID][4][16][2];    // paired B rows {2g, 2g+1} per col

  const int n = blockIdx.x;
  const int b0 = blockIdx.y * BPB;
  const int tid = threadIdx.x;

  // stage middle-site weights, pre-paired for float2 B-operand loads:
  // BmP[j][g][col] = { Wm[n,j,2g,  p(col), e(col)], Wm[n,j,2g+1, p(col), e(col)] }
  for (int i = tid; i < NMID * 4 * 16; i += 128) {
    int j = i >> 6, gg = (i >> 4) & 3, col = i & 15;
    int p = col >> 3, e = col & 7;
    const float* w = Wm + (n * NMID + j) * 128;
    BmP[j][gg][col][0] = w[((2 * gg) * 2 + p) * 8 + e];
    BmP[j][gg][col][1] = w[((2 * gg + 1) * 2 + p) * 8 + e];
  }
  // trig embedding of all site values
  for (int i = tid; i < BPB * S; i += 128) {
    int bb = i / S, s = i % S;
    float a = PI_HALF * v[((b0 + bb) * NPIX + n) * S + s];
    float sn, cs;
    __sincosf(a, &sn, &cs);
    csv[bb][s][0] = cs;
    csv[bb][s][1] = sn;
  }
  // init state: u0 = Wf(p=0), u1 = Wf(p=1)  (site-0 scale applied at first A-read)
  for (int i = tid; i < TILES * 256; i += 128) {
    int t = i >> 8, k = (i >> 4) & 15, m = i & 15;
    int o = (t * 16 + m) % O;
    uBuf[t][k][m] = Wf[((n * 2 + (k >> 3)) * O + o) * 8 + (k & 7)];
  }
  __syncthreads();

  const int lane = tid & 31, wave = tid >> 5;
  const int m = lane & 15;    // A row owned by this lane
  const int kk = lane >> 4;   // K-half selector
  const int col = lane & 15;  // B/D column owned by this lane

  for (int t = wave; t < TILES; t += 4) {
    const int row = t * 16 + m;
    const int bb = row / O;
    for (int j = 0; j < NMID; ++j) {
      v2f cs = *(const v2f*)&csv[bb][j][0];  // (cos, sin) for this lane's row
      v2f a0, a1;
      a0.x = uBuf[t][2 * kk][m]     * cs.x + uBuf[t][8 + 2 * kk][m]  * cs.y;
      a0.y = uBuf[t][2 * kk + 1][m] * cs.x + uBuf[t][9 + 2 * kk][m]  * cs.y;
      a1.x = uBuf[t][4 + 2 * kk][m] * cs.x + uBuf[t][12 + 2 * kk][m] * cs.y;
      a1.y = uBuf[t][5 + 2 * kk][m] * cs.x + uBuf[t][13 + 2 * kk][m] * cs.y;
      v2f bv0 = *(const v2f*)&BmP[j][kk][col][0];
      v2f bv1 = *(const v2f*)&BmP[j][2 + kk][col][0];
      v8f acc = {};
      acc = __builtin_amdgcn_wmma_f32_16x16x4_f32(false, a0, false, bv0,
                                                  (short)0, acc, false, false);
      acc = __builtin_amdgcn_wmma_f32_16x16x4_f32(false, a1, false, bv1,
                                                  (short)0, acc, false, false);
      // store D transposed: lane (col) covers m = 8*kk .. 8*kk+7, contiguous
      v4f lo = {acc[0], acc[1], acc[2], acc[3]};
      v4f hi = {acc[4], acc[5], acc[6], acc[7]};
      *(v4f*)&uBuf[t][col][8 * kk] = lo;
      *(v4f*)&uBuf[t][col][8 * kk + 4] = hi;
    }
    // epilogue: apply last middle-site scale (site S-2), contract with Wl (site S-1)
    if (lane < 16) {
      int o = row % O;
      float csM = csv[bb][S - 2][0], snM = csv[bb][S - 2][1];
      float csL = csv[bb][S - 1][0], snL = csv[bb][S - 1][1];
      const float* wl = Wl + n * 16;
      float accv = 0.0f;
#pragma unroll
      for (int d = 0; d < 8; ++d) {
        float cd = uBuf[t][d][m] * csM + uBuf[t][8 + d][m] * snM;
        accv += cd * (wl[2 * d] * csL + wl[2 * d + 1] * snL);
      }
      int b = b0 + bb;
      if (FINAL) {
        y[b * O + o] = accv;
      } else {
        y[(b * NPIX + n) * O + o] = accv;
        int ch = (n * O + o) >= half_thr ? 1 : 0;
        atomicAdd(&stats[ch], accv);
        atomicAdd(&stats[2 + ch], accv * accv);
      }
    }
  }
}

// ---------------------------------------------------------------------------
extern "C" void kernel_launch(void* const* d_in, const int* in_sizes, int n_in,
                              void* d_out, int out_size, void* d_ws, size_t ws_size,
                              hipStream_t stream) {
  (void)in_sizes; (void)n_in; (void)out_size; (void)ws_size;
  const float* x   = (const float*)d_in[0];
  const float* W0f = (const float*)d_in[1];
  const float* W0m = (const float*)d_in[2];
  const float* W0l = (const float*)d_in[3];
  const float* g0  = (const float*)d_in[4];
  const float* b0  = (const float*)d_in[5];
  const float* W1f = (const float*)d_in[6];
  const float* W1m = (const float*)d_in[7];
  const float* W1l = (const float*)d_in[8];
  const float* g1  = (const float*)d_in[9];
  const float* b1  = (const float*)d_in[10];
  const float* Wff = (const float*)d_in[11];
  const float* Wfm = (const float*)d_in[12];
  const float* Wfl = (const float*)d_in[13];
  float* out = (float*)d_out;

  float* ws  = (float*)d_ws;
  float* v0  = ws;               // 128*256*16 = 524288
  float* y0  = v0 + 524288;      // 128*256*2  =  65536
  float* st0 = y0 + 65536;       // 4
  float* v1  = st0 + 4;          // 128*16*32  =  65536
  float* y1  = v1 + 65536;       // 128*16*2   =   4096
  float* st1 = y1 + 4096;        // 4
  float* v2  = st1 + 4;          // 128*32     =   4096

  prep0_kernel<<<2048, 256, 0, stream>>>(x, v0, st0);
  mpo_chain_kernel<2, 64, 16, 14, 256, false>
      <<<dim3(256, 2), 128, 0, stream>>>(v0, W0f, W0m, W0l, y0, st0, 256);
  prep1_kernel<<<256, 256, 0, stream>>>(y0, st0, g0, b0, v1, st1);
  mpo_chain_kernel<2, 64, 32, 30, 16, false>
      <<<dim3(16, 2), 128, 0, stream>>>(v1, W1f, W1m, W1l, y1, st1, 16);
  prep2_kernel<<<16, 256, 0, stream>>>(y1, st1, g1, b1, v2);
  mpo_chain_kernel<10, 16, 32, 30, 1, true>
      <<<dim3(1, 8), 128, 0, stream>>>(v2, Wff, Wfm, Wfl, out, nullptr, 0);
}